// RBS_VQC_state_vector_31963146617522
// MI455X (gfx1250) — compile-verified
//
#include <hip/hip_runtime.h>
#include <stdint.h>

// Problem constants (from reference): N_QUBITS=16, HW_K=8 -> C(16,8)=12870 states,
// 32 layers of brickwork gates -> 16*8 + 16*7 = 240 gates, 3432 pairs per gate, batch 256.
#define N_STATES   12870
#define N_GATES    240
#define N_PAIRS    3432
#define IDX_INTS   (N_PAIRS * 2)          // 6864 ints per gate
#define IDX_VEC4   (IDX_INTS / 4)         // 1716 int4 per gate (6864 % 4 == 0)
#define BLOCK      256
#define VEC4_PER_THREAD 7                 // ceil(1716/256) -> fixed issue count per wave
#define IDX_PAD_VEC4 (VEC4_PER_THREAD * BLOCK)  // 1792 int4 slots per LDS buffer

// ---- CDNA5 async global<->LDS helpers (gfx1250, tracked by ASYNCcnt) -------
// LDS byte offset of a __shared__ object: flat shared address low 32 bits are the
// LDS offset (aperture bits live in addr[63:32] per CDNA5 ISA aperture rules).
__device__ __forceinline__ uint32_t lds_off(const void* p) {
  return (uint32_t)(uintptr_t)p;
}

__device__ __forceinline__ void async_load_lds_b128(uint32_t lds, const void* gaddr) {
  // GV mode: per-lane 64-bit global address, per-lane LDS byte address in VGPR.
  asm volatile("global_load_async_to_lds_b128 %0, %1, off"
               :: "v"(lds), "v"(gaddr) : "memory");
}
__device__ __forceinline__ void async_load_lds_b64(uint32_t lds, const void* gaddr) {
  asm volatile("global_load_async_to_lds_b64 %0, %1, off"
               :: "v"(lds), "v"(gaddr) : "memory");
}
__device__ __forceinline__ void async_store_lds_b64(const void* gaddr, uint32_t lds) {
  asm volatile("global_store_async_from_lds_b64 %0, %1, off"
               :: "v"(gaddr), "v"(lds) : "memory");
}
__device__ __forceinline__ void wait_async_le7() {
  asm volatile("s_wait_asynccnt 7" ::: "memory");
}
__device__ __forceinline__ void wait_async_0() {
  asm volatile("s_wait_asynccnt 0" ::: "memory");
}

// ---------------------------------------------------------------------------
// One workgroup owns one batch row, resident in LDS for all 240 gates.
// Index tables for gate g+1 are async-prefetched into a double buffer while
// gate g computes; every thread issues exactly VEC4_PER_THREAD b128 async
// loads (clamped tail) so every wave's ASYNCcnt advances by exactly 7 per
// gate, making `s_wait_asynccnt 7` retire exactly the current gate's data.
// ---------------------------------------------------------------------------
__global__ __launch_bounds__(BLOCK) void rbs_vqc_kernel(
    const float* __restrict__ input_state,   // [256][12870]
    const float* __restrict__ angles,        // [240]
    const int*   __restrict__ pairs,         // [240][3432][2]
    float*       __restrict__ out)           // [256][12870]
{
  __shared__ float  s_state[N_STATES];                 // 51,480 B
  __shared__ int    s_idx[2][IDX_PAD_VEC4 * 4];        // 2 * 28,672 B
  __shared__ float2 s_trig[N_GATES];                   // 1,920 B

  const int tid = threadIdx.x;
  const int b   = blockIdx.x;

  // ---- 1) async-load this batch row into LDS (rows are 8B aligned: 51480*b % 8 == 0)
  const float* src = input_state + (size_t)b * N_STATES;
  for (int j = tid; j < N_STATES / 2; j += BLOCK) {     // 6435 float2, exact cover
    async_load_lds_b64(lds_off(&s_state[2 * j]), (const void*)(src + 2 * j));
  }

  // ---- 2) async-prefetch gate 0 index table (7 b128 per thread, clamped tail)
  {
    const int* gbase = pairs;                // gate 0
    int* dst = s_idx[0];
#pragma unroll
    for (int k = 0; k < VEC4_PER_THREAD; ++k) {
      int j  = tid + k * BLOCK;
      int jg = (j < IDX_VEC4) ? j : (IDX_VEC4 - 1);     // clamp global side
      async_load_lds_b128(lds_off(&dst[4 * j]), (const void*)(gbase + 4 * jg));
    }
  }

  // ---- 3) trig table (one gate per thread; 240 <= 256)
  if (tid < N_GATES) {
    float th = angles[tid];
    s_trig[tid] = make_float2(cosf(th), sinf(th));
  }

  // ---- 4) gate chain, LDS-resident
  for (int g = 0; g < N_GATES; ++g) {
    if (g + 1 < N_GATES) {
      // prefetch next gate's indices into the other buffer
      const int* gbase = pairs + (size_t)(g + 1) * IDX_INTS;
      int* dst = s_idx[(g + 1) & 1];
#pragma unroll
      for (int k = 0; k < VEC4_PER_THREAD; ++k) {
        int j  = tid + k * BLOCK;
        int jg = (j < IDX_VEC4) ? j : (IDX_VEC4 - 1);
        async_load_lds_b128(lds_off(&dst[4 * j]), (const void*)(gbase + 4 * jg));
      }
      // retire everything issued before this prefetch (state row + gate g indices)
      wait_async_le7();
    } else {
      wait_async_0();
    }
    __syncthreads();   // make async LDS writes visible workgroup-wide

    const int*  idx = s_idx[g & 1];
    const float c = s_trig[g].x;
    const float s = s_trig[g].y;

    for (int p = tid; p < N_PAIRS; p += BLOCK) {
      const int2 ab = *(const int2*)(idx + 2 * p);      // ds_load_b64
      const float xi = s_state[ab.x];
      const float xj = s_state[ab.y];
      s_state[ab.x] = __builtin_fmaf(c, xi,  s * xj);
      s_state[ab.y] = __builtin_fmaf(c, xj, -s * xi);
    }
    __syncthreads();   // pairs of next gate may alias this gate's outputs
  }

  // ---- 5) async write-back LDS -> global
  float* dst = out + (size_t)b * N_STATES;
  for (int j = tid; j < N_STATES / 2; j += BLOCK) {
    async_store_lds_b64((const void*)(dst + 2 * j), lds_off(&s_state[2 * j]));
  }
  wait_async_0();
}

extern "C" void kernel_launch(void* const* d_in, const int* in_sizes, int n_in,
                              void* d_out, int out_size, void* d_ws, size_t ws_size,
                              hipStream_t stream) {
  const float* input_state = (const float*)d_in[0];   // 256*12870 f32
  const float* angles      = (const float*)d_in[1];   // 240 f32
  const int*   pairs       = (const int*)d_in[2];     // 240*3432*2 i32
  float*       out         = (float*)d_out;           // 256*12870 f32

  (void)in_sizes; (void)n_in; (void)out_size; (void)d_ws; (void)ws_size;

  rbs_vqc_kernel<<<256, BLOCK, 0, stream>>>(input_state, angles, pairs, out);
}